// DDECModel_84318797955381
// MI455X (gfx1250) — compile-verified
//
#include <hip/hip_runtime.h>
#include <hip/hip_bf16.h>
#include <math.h>

#define F_DIM 2048
#define E_DIM 4096
#define H_DIM 20
#define NEWTON_ITERS 2
#define EPS_CONST 0.1f
#define NB 64  // LU block size

typedef __attribute__((ext_vector_type(2))) float v2f;
typedef __attribute__((ext_vector_type(8))) float v8f;

static __device__ __forceinline__ v8f wmma_f32_16x16x4(v2f a, v2f b, v8f c) {
  // V_WMMA_F32_16X16X4_F32 : D = A(16x4) * B(4x16) + C(16x16), wave32
  return __builtin_amdgcn_wmma_f32_16x16x4_f32(false, a, false, b, (short)0, c,
                                               false, false);
}

// ---------------------------------------------------------------------------
// Small elementwise precompute: diagonal Hodge-star scalings.
// ---------------------------------------------------------------------------
__global__ __launch_bounds__(256) void precompute_kernel(
    const float* __restrict__ B1, const float* __restrict__ D1,
    const float* __restrict__ B2, const float* __restrict__ D2,
    float* __restrict__ w0, float* __restrict__ b1inv, float* __restrict__ d1inv,
    float* __restrict__ B2sq, float* __restrict__ D2sq) {
  int i = blockIdx.x * blockDim.x + threadIdx.x;
  if (i < E_DIM) {
    float bi = 1.0f / (B1[i] * B1[i] + 1e-5f);
    float di = 1.0f / (D1[i] * D1[i] + 1e-5f);
    b1inv[i] = bi;
    d1inv[i] = di;
    w0[i] = bi * di;
  }
  if (i < F_DIM) {
    B2sq[i] = B2[i] * B2[i];
    D2sq[i] = D2[i] * D2[i];
  }
}

__global__ __launch_bounds__(256) void weff_kernel(
    const float* __restrict__ w0, const float* __restrict__ dnn,
    float* __restrict__ weff) {
  int e = blockIdx.x * blockDim.x + threadIdx.x;
  if (e < E_DIM) weff[e] = w0[e] * (1.0f + EPS_CONST * dnn[e]);
}

__global__ __launch_bounds__(256) void axpy_update_kernel(
    float* __restrict__ u, const float* __restrict__ d) {
  int i = blockIdx.x * blockDim.x + threadIdx.x;
  if (i < F_DIM) u[i] -= d[i];
}

// ---------------------------------------------------------------------------
// Core WMMA GEMM, exploiting symmetry of S = (d1 diag(w)) d1^T:
//   C = diag(rowsc) * S * diag(colsc) + 1e-6 I
// Only the 136 lower-triangle 128x128 tiles are computed (1.88x FLOP saving);
// off-diagonal tiles are written in both orientations with swapped scalings.
// Block tile 128x128, 4 waves (2x2), wave tile 64x64 (4x4 WMMA tiles), BK=32.
// ---------------------------------------------------------------------------
__global__ __launch_bounds__(128) void build_sym_gemm(
    const float* __restrict__ d1, const float* __restrict__ weff,
    const float* __restrict__ rowsc, const float* __restrict__ colsc,
    float* __restrict__ C) {
  __shared__ float As[128][33];   // padded: conflict-free frag reads
  __shared__ float Bs[32][132];

  // decode lower-triangle tile (bi >= bj) from linear block id
  int t = blockIdx.x;
  int bi = (int)((sqrtf(8.0f * (float)t + 1.0f) - 1.0f) * 0.5f);
  while ((bi + 1) * (bi + 2) / 2 <= t) ++bi;
  while (bi * (bi + 1) / 2 > t) --bi;
  int bj = t - bi * (bi + 1) / 2;

  const int tid = threadIdx.x;
  const int lane = tid & 31;
  const int wave = tid >> 5;
  const int wm = (wave >> 1) * 64;
  const int wn = (wave & 1) * 64;
  const int mBase = bi * 128;
  const int nBase = bj * 128;
  const int lh = lane & 15;
  const int sel = lane >> 4;  // 0: lanes 0-15, 1: lanes 16-31

  v8f acc[4][4];
#pragma unroll
  for (int m = 0; m < 4; ++m)
#pragma unroll
    for (int n = 0; n < 4; ++n) acc[m][n] = (v8f){0, 0, 0, 0, 0, 0, 0, 0};

  for (int e0 = 0; e0 < E_DIM; e0 += 32) {
    // A tile: rows of d1 scaled by weff (folded; no materialized 32MB temp)
#pragma unroll
    for (int p = 0; p < 8; ++p) {
      int s = tid + p * 128;           // float4 slot 0..1023
      int r = s >> 3;
      int k4 = (s & 7) << 2;
      const float4 v = *reinterpret_cast<const float4*>(
          &d1[(size_t)(mBase + r) * E_DIM + e0 + k4]);
      const float4 wv = *reinterpret_cast<const float4*>(&weff[e0 + k4]);
      As[r][k4 + 0] = v.x * wv.x;
      As[r][k4 + 1] = v.y * wv.y;
      As[r][k4 + 2] = v.z * wv.z;
      As[r][k4 + 3] = v.w * wv.w;
    }
    // B tile: rows of d1 stored transposed into LDS (B = d1^T)
#pragma unroll
    for (int p = 0; p < 8; ++p) {
      int s = tid + p * 128;
      int n = s >> 3;
      int k4 = (s & 7) << 2;
      const float4 v = *reinterpret_cast<const float4*>(
          &d1[(size_t)(nBase + n) * E_DIM + e0 + k4]);
      Bs[k4 + 0][n] = v.x;
      Bs[k4 + 1][n] = v.y;
      Bs[k4 + 2][n] = v.z;
      Bs[k4 + 3][n] = v.w;
    }
    __syncthreads();

#pragma unroll
    for (int ks = 0; ks < 32; ks += 4) {
      v2f af[4], bf[4];
#pragma unroll
      for (int m = 0; m < 4; ++m) {
        int row = wm + m * 16 + lh;
        af[m].x = As[row][ks + 2 * sel + 0];  // ISA A layout: v0={K0,K2}
        af[m].y = As[row][ks + 2 * sel + 1];  //               v1={K1,K3}
      }
#pragma unroll
      for (int n = 0; n < 4; ++n) {
        int col = wn + n * 16 + lh;
        bf[n].x = Bs[ks + 2 * sel + 0][col];
        bf[n].y = Bs[ks + 2 * sel + 1][col];
      }
#pragma unroll
      for (int m = 0; m < 4; ++m)
#pragma unroll
        for (int n = 0; n < 4; ++n)
          acc[m][n] = wmma_f32_16x16x4(af[m], bf[n], acc[m][n]);
    }
    __syncthreads();
  }

  // Epilogue: diagonal scalings + 1e-6 I; mirror write for off-diag tiles
#pragma unroll
  for (int m = 0; m < 4; ++m) {
#pragma unroll
    for (int n = 0; n < 4; ++n) {
      int col = nBase + wn + n * 16 + lh;
      float csc = colsc[col];
      float rsc_c = rowsc[col];
#pragma unroll
      for (int r = 0; r < 8; ++r) {
        int row = mBase + wm + m * 16 + sel * 8 + r;  // C layout: vr = r, r+8
        float s = acc[m][n][r];
        float v = s * rowsc[row] * csc;
        if (row == col) v += 1e-6f;
        C[(size_t)row * F_DIM + col] = v;
        if (bi != bj) C[(size_t)col * F_DIM + row] = s * rsc_c * colsc[row];
      }
    }
  }
}

// ---------------------------------------------------------------------------
// Blocked unpivoted LU (NB=64). Four kernels per step; trailing update is the
// FLOP-dominant part and uses WMMA (K=64).
// ---------------------------------------------------------------------------
__global__ __launch_bounds__(64) void lu_diag_kernel(float* __restrict__ A,
                                                     int k0) {
  __shared__ float Ds[64][65];
  int tid = threadIdx.x;
  for (int i = 0; i < 64; ++i)
    Ds[i][tid] = A[(size_t)(k0 + i) * F_DIM + k0 + tid];
  __syncthreads();
  for (int c = 0; c < 64; ++c) {
    if (tid > c) {
      float l = Ds[tid][c] / Ds[c][c];
      Ds[tid][c] = l;
      for (int cc = c + 1; cc < 64; ++cc) Ds[tid][cc] -= l * Ds[c][cc];
    }
    __syncthreads();
  }
  for (int i = 0; i < 64; ++i)
    A[(size_t)(k0 + i) * F_DIM + k0 + tid] = Ds[i][tid];
}

// L21 = A21 * U11^-1  (64 rows per workgroup, solved row-wise in LDS)
__global__ __launch_bounds__(64) void lu_left_kernel(float* __restrict__ A,
                                                     int k0) {
  __shared__ float Us[64][65];
  __shared__ float Xs[64][65];
  int tid = threadIdx.x;
  int rBase = k0 + NB + blockIdx.x * 64;
  for (int i = 0; i < 64; ++i) {
    Us[i][tid] = A[(size_t)(k0 + i) * F_DIM + k0 + tid];
    Xs[i][tid] = A[(size_t)(rBase + i) * F_DIM + k0 + tid];
  }
  __syncthreads();
  // thread tid owns row tid: solve x * U11 = a
  for (int c = 0; c < 64; ++c) {
    float x = Xs[tid][c];
    for (int t2 = 0; t2 < c; ++t2) x -= Xs[tid][t2] * Us[t2][c];
    Xs[tid][c] = x / Us[c][c];
  }
  __syncthreads();
  for (int i = 0; i < 64; ++i)
    A[(size_t)(rBase + i) * F_DIM + k0 + tid] = Xs[i][tid];
}

// U12 = L11^-1 * A12  (64 cols per workgroup, thread per column)
__global__ __launch_bounds__(64) void lu_right_kernel(float* __restrict__ A,
                                                      int k0) {
  __shared__ float Ls[64][65];
  __shared__ float Xs[64][65];
  int tid = threadIdx.x;
  int cBase = k0 + NB + blockIdx.x * 64;
  for (int i = 0; i < 64; ++i) {
    Ls[i][tid] = A[(size_t)(k0 + i) * F_DIM + k0 + tid];
    Xs[i][tid] = A[(size_t)(k0 + i) * F_DIM + cBase + tid];
  }
  __syncthreads();
  for (int r = 0; r < 64; ++r) {
    float x = Xs[r][tid];
    for (int t2 = 0; t2 < r; ++t2) x -= Ls[r][t2] * Xs[t2][tid];
    Xs[r][tid] = x;  // unit-diag L
  }
  __syncthreads();
  for (int i = 0; i < 64; ++i)
    A[(size_t)(k0 + i) * F_DIM + cBase + tid] = Xs[i][tid];
}

// A22 -= L21 * U12 : WMMA GEMM, 64x64 tile per WG, 4 waves (2x2), wave 32x32
__global__ __launch_bounds__(128) void lu_trail_kernel(float* __restrict__ A,
                                                       int k0) {
  __shared__ float Ls[64][68];
  __shared__ float Us[64][68];
  const int k2 = k0 + NB;
  const int tid = threadIdx.x;
  const int lane = tid & 31;
  const int wave = tid >> 5;
  const int wm = (wave >> 1) * 32;
  const int wn = (wave & 1) * 32;
  const int rBase = k2 + blockIdx.y * 64;
  const int cBase = k2 + blockIdx.x * 64;
  const int lh = lane & 15;
  const int sel = lane >> 4;

#pragma unroll
  for (int p = 0; p < 8; ++p) {
    int s = tid + p * 128;  // float4 slot 0..1023
    int r = s >> 4;
    int k4 = (s & 15) << 2;
    float4 va = *reinterpret_cast<const float4*>(
        &A[(size_t)(rBase + r) * F_DIM + k0 + k4]);
    Ls[r][k4 + 0] = va.x; Ls[r][k4 + 1] = va.y;
    Ls[r][k4 + 2] = va.z; Ls[r][k4 + 3] = va.w;
    float4 vb = *reinterpret_cast<const float4*>(
        &A[(size_t)(k0 + r) * F_DIM + cBase + k4]);
    Us[r][k4 + 0] = vb.x; Us[r][k4 + 1] = vb.y;
    Us[r][k4 + 2] = vb.z; Us[r][k4 + 3] = vb.w;
  }
  __syncthreads();

  v8f acc[2][2];
#pragma unroll
  for (int m = 0; m < 2; ++m)
#pragma unroll
    for (int n = 0; n < 2; ++n) acc[m][n] = (v8f){0, 0, 0, 0, 0, 0, 0, 0};

#pragma unroll
  for (int ks = 0; ks < 64; ks += 4) {
    v2f af[2], bf[2];
#pragma unroll
    for (int m = 0; m < 2; ++m) {
      int row = wm + m * 16 + lh;
      af[m].x = Ls[row][ks + 2 * sel + 0];
      af[m].y = Ls[row][ks + 2 * sel + 1];
    }
#pragma unroll
    for (int n = 0; n < 2; ++n) {
      int col = wn + n * 16 + lh;
      bf[n].x = Us[ks + 2 * sel + 0][col];
      bf[n].y = Us[ks + 2 * sel + 1][col];
    }
#pragma unroll
    for (int m = 0; m < 2; ++m)
#pragma unroll
      for (int n = 0; n < 2; ++n)
        acc[m][n] = wmma_f32_16x16x4(af[m], bf[n], acc[m][n]);
  }

#pragma unroll
  for (int m = 0; m < 2; ++m)
#pragma unroll
    for (int n = 0; n < 2; ++n) {
      int col = cBase + wn + n * 16 + lh;
#pragma unroll
      for (int r = 0; r < 8; ++r) {
        int row = rBase + wm + m * 16 + sel * 8 + r;
        A[(size_t)row * F_DIM + col] -= acc[m][n][r];
      }
    }
}

// ---------------------------------------------------------------------------
// RHS triangular solve: L y = b (unit lower), then U x = y. Single WG,
// blocked by 64; diag blocks solved in-wave with shuffles (no barriers).
// ---------------------------------------------------------------------------
__global__ __launch_bounds__(1024) void lu_solve_kernel(
    const float* __restrict__ A, const float* __restrict__ b,
    float* __restrict__ x) {
  __shared__ float yl[F_DIM];
  __shared__ float Ds[64][65];
  int tid = threadIdx.x;
  for (int i = tid; i < F_DIM; i += 1024) yl[i] = b[i];
  __syncthreads();

  // forward
  for (int kb = 0; kb < F_DIM / NB; ++kb) {
    int k0 = kb * NB;
    for (int idx = tid; idx < 64 * 64; idx += 1024) {
      int r = idx >> 6, c = idx & 63;
      Ds[r][c] = A[(size_t)(k0 + r) * F_DIM + k0 + c];
    }
    __syncthreads();
    if (tid < 32) {
      float y0 = yl[k0 + tid], y1 = yl[k0 + 32 + tid];
      for (int c = 0; c < 64; ++c) {
        float yc = (c < 32) ? __shfl(y0, c, 32) : __shfl(y1, c - 32, 32);
        if (c < 32) {
          if (tid > c) y0 -= Ds[tid][c] * yc;
          y1 -= Ds[32 + tid][c] * yc;
        } else {
          if (tid > c - 32) y1 -= Ds[32 + tid][c] * yc;
        }
      }
      yl[k0 + tid] = y0;
      yl[k0 + 32 + tid] = y1;
    }
    __syncthreads();
    for (int r = k0 + 64 + tid; r < F_DIM; r += 1024) {
      float acc = yl[r];
      const float* Ar = &A[(size_t)r * F_DIM + k0];
#pragma unroll 8
      for (int c = 0; c < 64; ++c) acc -= Ar[c] * yl[k0 + c];
      yl[r] = acc;
    }
    __syncthreads();
  }

  // backward
  for (int kb = F_DIM / NB - 1; kb >= 0; --kb) {
    int k0 = kb * NB;
    for (int idx = tid; idx < 64 * 64; idx += 1024) {
      int r = idx >> 6, c = idx & 63;
      Ds[r][c] = A[(size_t)(k0 + r) * F_DIM + k0 + c];
    }
    __syncthreads();
    if (tid < 32) {
      float y0 = yl[k0 + tid], y1 = yl[k0 + 32 + tid];
      for (int c = 63; c >= 0; --c) {
        if (c >= 32) {
          int cc = c - 32;
          if (tid == cc) y1 = y1 / Ds[c][c];
          float xc = __shfl(y1, cc, 32);
          if (tid < cc) y1 -= Ds[32 + tid][c] * xc;
          y0 -= Ds[tid][c] * xc;
        } else {
          if (tid == c) y0 = y0 / Ds[c][c];
          float xc = __shfl(y0, c, 32);
          if (tid < c) y0 -= Ds[tid][c] * xc;
        }
      }
      yl[k0 + tid] = y0;
      yl[k0 + 32 + tid] = y1;
    }
    __syncthreads();
    for (int r = tid; r < k0; r += 1024) {
      float acc = yl[r];
      const float* Ar = &A[(size_t)r * F_DIM + k0];
#pragma unroll 8
      for (int c = 0; c < 64; ++c) acc -= Ar[c] * yl[k0 + c];
      yl[r] = acc;
    }
    __syncthreads();
  }
  for (int i = tid; i < F_DIM; i += 1024) x[i] = yl[i];
}

// ---------------------------------------------------------------------------
// gu = d1inv .* (d1^T @ (D2sq .* u))  : thread-per-edge, coalesced along e
// ---------------------------------------------------------------------------
__global__ __launch_bounds__(256) void grad_matvec_kernel(
    const float* __restrict__ d1, const float* __restrict__ D2sq,
    const float* __restrict__ u, const float* __restrict__ d1inv,
    float* __restrict__ gu) {
  int e = blockIdx.x * blockDim.x + threadIdx.x;
  float acc = 0.0f;
  for (int f = 0; f < F_DIM; ++f)
    acc += d1[(size_t)f * E_DIM + e] * (D2sq[f] * u[f]);
  gu[e] = acc * d1inv[e];
}

// ---------------------------------------------------------------------------
// Per-edge 1->20->20->1 MLP: value + exact derivative wrt scalar input (jvp).
// ---------------------------------------------------------------------------
__global__ __launch_bounds__(256) void mlp_kernel(
    const float* __restrict__ gu, const float* __restrict__ W1,
    const float* __restrict__ b1, const float* __restrict__ W2,
    const float* __restrict__ b2, const float* __restrict__ W3,
    const float* __restrict__ b3, float* __restrict__ val,
    float* __restrict__ dnn) {
  int e = blockIdx.x * blockDim.x + threadIdx.x;
  if (e >= E_DIM) return;
  float g = gu[e];
  float h[H_DIM], hd[H_DIM];
#pragma unroll
  for (int i = 0; i < H_DIM; ++i) {
    float w = W1[e * H_DIM + i];
    float z = g * w + b1[e * H_DIM + i];
    float ez = __expf(z);
    bool pos = z > 0.0f;
    h[i] = pos ? z : (ez - 1.0f);      // elu
    hd[i] = (pos ? 1.0f : ez) * w;     // d h / d g
  }
  float vacc = 0.0f, dacc = 0.0f;
  for (int i = 0; i < H_DIM; ++i) {
    float a = b2[e * H_DIM + i], ad = 0.0f;
    const float* w2row = &W2[(size_t)e * H_DIM * H_DIM + i * H_DIM];
#pragma unroll
    for (int j = 0; j < H_DIM; ++j) {
      float w = w2row[j];
      a += w * h[j];
      ad += w * hd[j];
    }
    float ez = __expf(a);
    bool pos = a > 0.0f;
    float h2 = pos ? a : (ez - 1.0f);
    float d2 = (pos ? 1.0f : ez) * ad;
    float w3 = W3[e * H_DIM + i];
    vacc += w3 * h2;
    dacc += w3 * d2;
  }
  val[e] = vacc + b3[e];
  dnn[e] = dacc;
}

// ---------------------------------------------------------------------------
// res = K u + EPS * B2sq .* (d1 @ (b1inv .* val)) - f ; wave-per-row
// ---------------------------------------------------------------------------
__global__ __launch_bounds__(256) void residual_kernel(
    const float* __restrict__ K, const float* __restrict__ u,
    const float* __restrict__ d1, const float* __restrict__ b1inv,
    const float* __restrict__ val, const float* __restrict__ B2sq,
    const float* __restrict__ f, float* __restrict__ res) {
  int lane = threadIdx.x & 31;
  int wave = threadIdx.x >> 5;
  int row = blockIdx.x * 8 + wave;
  const float* Kr = &K[(size_t)row * F_DIM];
  float acc = 0.0f;
  for (int j = lane; j < F_DIM; j += 32) acc += Kr[j] * u[j];
  const float* dr = &d1[(size_t)row * E_DIM];
  float acc2 = 0.0f;
  for (int e = lane; e < E_DIM; e += 32) acc2 += dr[e] * b1inv[e] * val[e];
#pragma unroll
  for (int off = 16; off > 0; off >>= 1) {
    acc += __shfl_xor(acc, off, 32);
    acc2 += __shfl_xor(acc2, off, 32);
  }
  if (lane == 0) res[row] = acc + EPS_CONST * B2sq[row] * acc2 - f[row];
}

// ---------------------------------------------------------------------------
extern "C" void kernel_launch(void* const* d_in, const int* in_sizes, int n_in,
                              void* d_out, int out_size, void* d_ws,
                              size_t ws_size, hipStream_t stream) {
  (void)in_sizes; (void)n_in; (void)out_size; (void)ws_size;
  const float* f_in = (const float*)d_in[1];   // d_in[0] = u (unused by ref)
  const float* d1 = (const float*)d_in[2];
  const float* B1v = (const float*)d_in[3];
  const float* B2v = (const float*)d_in[4];
  const float* D1v = (const float*)d_in[5];
  const float* D2v = (const float*)d_in[6];
  const float* W1 = (const float*)d_in[7];
  const float* b1p = (const float*)d_in[8];
  const float* W2 = (const float*)d_in[9];
  const float* b2p = (const float*)d_in[10];
  const float* W3 = (const float*)d_in[11];
  const float* b3p = (const float*)d_in[12];

  // workspace layout (floats): ~33.7 MB total
  float* ws = (float*)d_ws;
  float* Kmat = ws;
  float* Amat = Kmat + (size_t)F_DIM * F_DIM;
  float* w0 = Amat + (size_t)F_DIM * F_DIM;
  float* b1inv = w0 + E_DIM;
  float* d1inv = b1inv + E_DIM;
  float* weff = d1inv + E_DIM;
  float* gu = weff + E_DIM;
  float* val = gu + E_DIM;
  float* dnn = val + E_DIM;
  float* B2sq = dnn + E_DIM;
  float* D2sq = B2sq + F_DIM;
  float* u_n = D2sq + F_DIM;
  float* res = u_n + F_DIM;
  float* dlt = res + F_DIM;

  auto run_lu = [&](float* M) {
    for (int kb = 0; kb < F_DIM / NB; ++kb) {
      int k0 = kb * NB;
      lu_diag_kernel<<<1, 64, 0, stream>>>(M, k0);
      int rem = F_DIM - k0 - NB;
      if (rem > 0) {
        lu_left_kernel<<<rem / 64, 64, 0, stream>>>(M, k0);
        lu_right_kernel<<<rem / 64, 64, 0, stream>>>(M, k0);
        dim3 g(rem / 64, rem / 64);
        lu_trail_kernel<<<g, 128, 0, stream>>>(M, k0);
      }
    }
  };

  precompute_kernel<<<E_DIM / 256, 256, 0, stream>>>(
      B1v, D1v, B2v, D2v, w0, b1inv, d1inv, B2sq, D2sq);

  // K = diag(B2^2) (d1 diag(w0) d1^T) diag(D2^2) + 1e-6 I
  const int NTB = F_DIM / 128;                 // 16 tile-rows
  const int TRI_BLOCKS = NTB * (NTB + 1) / 2;  // 136 lower-triangle tiles
  build_sym_gemm<<<TRI_BLOCKS, 128, 0, stream>>>(d1, w0, B2sq, D2sq, Kmat);

  // u_n = K^-1 f
  hipMemcpyAsync(Amat, Kmat, (size_t)F_DIM * F_DIM * sizeof(float),
                 hipMemcpyDeviceToDevice, stream);
  run_lu(Amat);
  lu_solve_kernel<<<1, 1024, 0, stream>>>(Amat, f_in, u_n);

  for (int it = 0; it < NEWTON_ITERS; ++it) {
    grad_matvec_kernel<<<E_DIM / 256, 256, 0, stream>>>(d1, D2sq, u_n, d1inv,
                                                        gu);
    mlp_kernel<<<E_DIM / 256, 256, 0, stream>>>(gu, W1, b1p, W2, b2p, W3, b3p,
                                                val, dnn);
    residual_kernel<<<F_DIM / 8, 256, 0, stream>>>(Kmat, u_n, d1, b1inv, val,
                                                   B2sq, f_in, res);
    // J = diag(B2^2) (d1 diag(w0*(1+eps*dnn)) d1^T) diag(D2^2) + 1e-6 I
    weff_kernel<<<E_DIM / 256, 256, 0, stream>>>(w0, dnn, weff);
    build_sym_gemm<<<TRI_BLOCKS, 128, 0, stream>>>(d1, weff, B2sq, D2sq, Amat);
    run_lu(Amat);
    lu_solve_kernel<<<1, 1024, 0, stream>>>(Amat, res, dlt);
    axpy_update_kernel<<<F_DIM / 256, 256, 0, stream>>>(u_n, dlt);
  }

  hipMemcpyAsync(d_out, u_n, F_DIM * sizeof(float), hipMemcpyDeviceToDevice,
                 stream);
}